// MoEWeighting_58789512348486
// MI455X (gfx1250) — compile-verified
//
#include <hip/hip_runtime.h>
#include <hip/hip_bf16.h>

typedef float v2f __attribute__((ext_vector_type(2)));
typedef float v4f __attribute__((ext_vector_type(4)));
typedef float v8f __attribute__((ext_vector_type(8)));

#define DFEAT 384
#define NEXP  3
#define PART_BLOCKS 64
// ws layout (floats): [0..383] agg column sums, [384..386] gate weights,
//                     [512 .. 512+64*384) per-block partial column sums
#define WS_GATE   384
#define WS_PART   512

// ---------------- Stage 1: partial column sums of query (deterministic) ----------------
__global__ __launch_bounds__(DFEAT) void qsum_partial(const float* __restrict__ q,
                                                      float* __restrict__ ws,
                                                      int m, int rows_per_block) {
    const int t  = threadIdx.x;              // column 0..383
    const int r0 = blockIdx.x * rows_per_block;
    int r1 = r0 + rows_per_block; if (r1 > m) r1 = m;
    float s = 0.f;
    for (int r = r0; r < r1; ++r) s += q[(size_t)r * DFEAT + t];
    ws[WS_PART + blockIdx.x * DFEAT + t] = s;
}

// ---------------- Stage 2: fold partials into column sums ----------------
__global__ __launch_bounds__(DFEAT) void qsum_fold(float* __restrict__ ws) {
    const int t = threadIdx.x;
    float s = 0.f;
    #pragma unroll 8
    for (int b = 0; b < PART_BLOCKS; ++b) s += ws[WS_PART + b * DFEAT + t];
    ws[t] = s;
}

// ---------------- Stage 3: gate via V_WMMA_F32_16X16X4_F32 (one wave) ----------------
// A (16x4): rows 0..2 = Wg[e][k0+kl]; B (4x16): every column = agg[k0+kl].
// C[e][0] accumulates Wg[e] . agg over the K loop.
__global__ __launch_bounds__(32) void gate_kernel(const float* __restrict__ ws_sums,
                                                  const float* __restrict__ Wg,
                                                  const float* __restrict__ bg,
                                                  float* __restrict__ gate_out,
                                                  float inv_m) {
    const int lane = threadIdx.x;
    const int m16  = lane & 15;
    const int kh   = (lane < 16) ? 0 : 2;          // 32-bit A/B layout: K halves split at lane 16
    const int row  = (m16 < NEXP) ? m16 : 0;       // clamp, then mask (branchless, EXEC stays full)
    const float msk = (m16 < NEXP) ? 1.f : 0.f;

    v8f c = {};
    for (int k0 = 0; k0 < DFEAT; k0 += 4) {
        v2f a, b;
        b[0] = ws_sums[k0 + kh]     * inv_m;
        b[1] = ws_sums[k0 + kh + 1] * inv_m;
        a[0] = Wg[row * DFEAT + k0 + kh]     * msk;
        a[1] = Wg[row * DFEAT + k0 + kh + 1] * msk;
        c = __builtin_amdgcn_wmma_f32_16x16x4_f32(false, a, false, b, (short)0, c,
                                                  false, false);
    }

    if (lane == 0) {
        float s0 = c[0] + bg[0];
        float s1 = c[1] + bg[1];
        float s2 = c[2] + bg[2];
        float mx = fmaxf(s0, fmaxf(s1, s2));
        float p0 = __expf(s0 - mx), p1 = __expf(s1 - mx), p2 = __expf(s2 - mx);
        float inv = 1.f / (p0 + p1 + p2);
        p0 *= inv; p1 *= inv; p2 *= inv;
        // drop the single smallest (top-2 of 3), renormalize
        float pmin = fminf(p0, fminf(p1, p2));
        if (p0 == pmin)      p0 = 0.f;
        else if (p1 == pmin) p1 = 0.f;
        else                 p2 = 0.f;
        float rn = 1.f / (p0 + p1 + p2);
        gate_out[0] = p0 * rn;
        gate_out[1] = p1 * rn;
        gate_out[2] = p2 * rn;
    }
}

// ---------------- Stage 4: bandwidth-bound streaming pass (one point per wave) ----------------
__global__ __launch_bounds__(256) void moe_main(const float* __restrict__ feat,
                                                const float* __restrict__ Wexp,
                                                const float* __restrict__ bexp,
                                                const float* __restrict__ gate,
                                                float* __restrict__ out, int n) {
    const int lane  = threadIdx.x & 31;
    const int point = (int)((blockIdx.x * (unsigned)blockDim.x + threadIdx.x) >> 5);
    if (point >= n) return;

    const float g0 = gate[0], g1 = gate[1], g2 = gate[2];
    const float b0 = bexp[0], b1 = bexp[1], b2 = bexp[2];

    const v4f* __restrict__ W4 = (const v4f*)Wexp;                       // 3*96 v4f, L2-resident
    const v4f* __restrict__ F4 = (const v4f*)feat + (size_t)point * (NEXP * DFEAT / 4);
    v4f*       __restrict__ O4 = (v4f*)out + (size_t)point * (DFEAT / 4);

    // Per-lane tiles: lane owns v4f slots {lane, lane+32, lane+64} of each 96-v4f row.
    v4f w[NEXP][3], f[NEXP][3];
    #pragma unroll
    for (int e = 0; e < NEXP; ++e)
        #pragma unroll
        for (int j = 0; j < 3; ++j) {
            w[e][j] = W4[e * 96 + 32 * j + lane];
            f[e][j] = __builtin_nontemporal_load(&F4[e * 96 + 32 * j + lane]); // no L2 reuse
        }

    // Per-expert partial dots, then wave32 butterfly reduction (all lanes get full sum).
    float dt[NEXP];
    #pragma unroll
    for (int e = 0; e < NEXP; ++e) {
        float s = 0.f;
        #pragma unroll
        for (int j = 0; j < 3; ++j)
            #pragma unroll
            for (int k = 0; k < 4; ++k)
                s = fmaf(f[e][j][k], w[e][j][k], s);
        dt[e] = s;
    }
    #pragma unroll
    for (int off = 16; off > 0; off >>= 1) {
        dt[0] += __shfl_xor(dt[0], off, 32);
        dt[1] += __shfl_xor(dt[1], off, 32);
        dt[2] += __shfl_xor(dt[2], off, 32);
    }

    // Gated scores (dropped expert -> exact 0, still in the softmax, matching reference).
    float s0 = (dt[0] + b0) * g0;
    float s1 = (dt[1] + b1) * g1;
    float s2 = (dt[2] + b2) * g2;
    float mx = fmaxf(s0, fmaxf(s1, s2));
    float p0 = __expf(s0 - mx), p1 = __expf(s1 - mx), p2 = __expf(s2 - mx);
    float inv = 1.f / (p0 + p1 + p2);
    p0 *= inv; p1 *= inv; p2 *= inv;

    #pragma unroll
    for (int j = 0; j < 3; ++j) {
        v4f o = p0 * f[0][j] + p1 * f[1][j] + p2 * f[2][j];
        __builtin_nontemporal_store(o, &O4[32 * j + lane]);   // write-once output
    }
}

extern "C" void kernel_launch(void* const* d_in, const int* in_sizes, int n_in,
                              void* d_out, int out_size, void* d_ws, size_t ws_size,
                              hipStream_t stream) {
    const float* query = (const float*)d_in[0];
    const float* feat  = (const float*)d_in[1];
    const float* Wexp  = (const float*)d_in[2];
    const float* bexp  = (const float*)d_in[3];
    const float* Wg    = (const float*)d_in[4];
    const float* bg    = (const float*)d_in[5];
    float* out = (float*)d_out;
    float* ws  = (float*)d_ws;

    const int m = in_sizes[0] / DFEAT;                 // 4096
    const int n = in_sizes[1] / (NEXP * DFEAT);        // 200000

    const int rpb = (m + PART_BLOCKS - 1) / PART_BLOCKS;
    qsum_partial<<<PART_BLOCKS, DFEAT, 0, stream>>>(query, ws, m, rpb);
    qsum_fold<<<1, DFEAT, 0, stream>>>(ws);
    gate_kernel<<<1, 32, 0, stream>>>(ws, Wg, bg, ws + WS_GATE, 1.0f / (float)m);

    const int blocks = (n + 7) / 8;                    // 8 waves/block, 1 point/wave
    moe_main<<<blocks, 256, 0, stream>>>(feat, Wexp, bexp, ws + WS_GATE, out, n);
}